// QoRNet_26328149525185
// MI455X (gfx1250) — compile-verified
//
#include <hip/hip_runtime.h>
#include <hip/hip_bf16.h>

typedef float v2f __attribute__((ext_vector_type(2)));
typedef float v8f __attribute__((ext_vector_type(8)));

#define HIDW   128
#define HEADS  4
#define CHV    32
#define LAYERS 3
#define NODE_DIM 64
#define EDGE_DIM 16
#define REC_DIM  8
#define KIN (NODE_DIM + REC_DIM)   // 72
#define GRAPHS 8

// ---- order-preserving float <-> uint key (for atomic segment-max) ----
__device__ __forceinline__ unsigned fkey(float f) {
    unsigned b = __float_as_uint(f);
    return (b & 0x80000000u) ? ~b : (b | 0x80000000u);
}
__device__ __forceinline__ float funkey(unsigned k) {
    unsigned b = (k & 0x80000000u) ? (k ^ 0x80000000u) : ~k;
    return __uint_as_float(b);
}

// =====================================================================
// Fold edge-attention weights:  W_att[k, l*4+h] = sum_c lin_edge_w[l][k,h*32+c]*att_edge[l,h,c]
// =====================================================================
__global__ void k_attw(const float* __restrict__ lin_edge_w,
                       const float* __restrict__ att_edge,
                       float* __restrict__ wattall) {
    int i = blockIdx.x * blockDim.x + threadIdx.x;    // k*12 + j
    if (i >= HIDW * 12) return;
    int k = i / 12, j = i % 12;
    int l = j >> 2, hh = j & 3;
    const float* wrow = lin_edge_w + (size_t)l * HIDW * HIDW + (size_t)k * HIDW + hh * CHV;
    const float* ae   = att_edge + (size_t)j * CHV;
    float s = 0.f;
    #pragma unroll
    for (int c = 0; c < CHV; c++) s += wrow[c] * ae[c];
    wattall[i] = s;
}

// Wc[16,12] = W_edge @ W_att ; bc[12] = b_edge @ W_att   (single block)
__global__ void k_wc(const float* __restrict__ W_edge, const float* __restrict__ b_edge,
                     const float* __restrict__ wattall,
                     float* __restrict__ wc, float* __restrict__ bcomb) {
    int i = threadIdx.x;
    if (i >= 17 * 12) return;
    int d = i / 12, j = i % 12;
    const float* row = (d < EDGE_DIM) ? (W_edge + (size_t)d * HIDW) : b_edge;
    float s = 0.f;
    for (int k = 0; k < HIDW; k++) s += row[k] * wattall[k * 12 + j];
    if (d < EDGE_DIM) wc[i] = s; else bcomb[j] = s;
}

// a_edge_all[E,12] = edge_attr @ Wc + bc
__global__ __launch_bounds__(256)
void k_aedge(const float* __restrict__ edge_attr, const float* __restrict__ wc,
             const float* __restrict__ bcomb, float* __restrict__ aedge_all, int E) {
    __shared__ float sW[EDGE_DIM * 12];
    __shared__ float sB[12];
    if (threadIdx.x < EDGE_DIM * 12) sW[threadIdx.x] = wc[threadIdx.x];
    if (threadIdx.x < 12) sB[threadIdx.x] = bcomb[threadIdx.x];
    __syncthreads();
    int e = blockIdx.x * blockDim.x + threadIdx.x;
    if (e >= E) return;
    float ea[EDGE_DIM];
    const float4* p = (const float4*)(edge_attr + (size_t)e * EDGE_DIM);
    #pragma unroll
    for (int q = 0; q < 4; q++) {
        float4 v = p[q];
        ea[q*4+0] = v.x; ea[q*4+1] = v.y; ea[q*4+2] = v.z; ea[q*4+3] = v.w;
    }
    #pragma unroll
    for (int j = 0; j < 12; j++) {
        float s = sB[j];
        #pragma unroll
        for (int d = 0; d < EDGE_DIM; d++) s += ea[d] * sW[d * 12 + j];
        aedge_all[(size_t)e * 12 + j] = s;
    }
}

// =====================================================================
// Input projection: h = relu(concat(x, recipe[batch]) @ W_in + b_in)
// f32 WMMA 16x16x4, block = 8 waves = 16-row x 128-col tile
// =====================================================================
__global__ __launch_bounds__(256)
void k_inproj(const float* __restrict__ x, const int* __restrict__ batch,
              const float* __restrict__ recipe, const float* __restrict__ W_in,
              const float* __restrict__ b_in, float* __restrict__ h, int N) {
    __shared__ __align__(16) float As[16][76];   // K=72, stride 76 -> conflict-free
    int n0 = blockIdx.x * 16;
    for (int idx = threadIdx.x; idx < 16 * KIN; idx += 256) {
        int r = idx / KIN, d = idx - r * KIN;
        int n = n0 + r;
        float v = 0.f;
        if (n < N) v = (d < NODE_DIM) ? x[(size_t)n * NODE_DIM + d]
                                      : recipe[(size_t)batch[n] * REC_DIM + (d - NODE_DIM)];
        As[r][d] = v;
    }
    __syncthreads();
    int lane = threadIdx.x & 31, wave = threadIdx.x >> 5;
    int m16 = lane & 15, kh = lane >> 4;
    int col = wave * 16 + m16;
    v8f c = {};
    #pragma unroll
    for (int k0 = 0; k0 < KIN; k0 += 4) {
        int ka = k0 + 2 * kh;
        v2f a, b;
        a.x = As[m16][ka];  a.y = As[m16][ka + 1];
        b.x = W_in[(size_t)ka * HIDW + col];
        b.y = W_in[(size_t)(ka + 1) * HIDW + col];
        c = __builtin_amdgcn_wmma_f32_16x16x4_f32(false, a, false, b, (short)0, c, false, false);
    }
    float bias = b_in[col];
    #pragma unroll
    for (int i = 0; i < 8; i++) {
        int n = n0 + i + kh * 8;
        if (n < N) {
            float v = c[i] + bias;
            h[(size_t)n * HIDW + col] = v > 0.f ? v : 0.f;
        }
    }
}

// =====================================================================
// xl = h @ W  (128x128), no activation.  f32 WMMA, 16x128 tile/block.
// =====================================================================
__global__ __launch_bounds__(256)
void k_lin(const float* __restrict__ h, const float* __restrict__ W,
           float* __restrict__ xl, int N) {
    __shared__ __align__(16) float As[16][132];  // stride 132 -> conflict-free
    int n0 = blockIdx.x * 16;
    for (int idx = threadIdx.x; idx < 16 * 32; idx += 256) {
        int r = idx >> 5, q = idx & 31;
        int n = n0 + r;
        float4 v = make_float4(0.f, 0.f, 0.f, 0.f);
        if (n < N) v = ((const float4*)(h + (size_t)n * HIDW))[q];
        ((float4*)&As[r][0])[q] = v;
    }
    __syncthreads();
    int lane = threadIdx.x & 31, wave = threadIdx.x >> 5;
    int m16 = lane & 15, kh = lane >> 4;
    int col = wave * 16 + m16;
    v8f c = {};
    #pragma unroll
    for (int k0 = 0; k0 < HIDW; k0 += 4) {
        int ka = k0 + 2 * kh;
        v2f a, b;
        a.x = As[m16][ka];  a.y = As[m16][ka + 1];
        b.x = W[(size_t)ka * HIDW + col];
        b.y = W[(size_t)(ka + 1) * HIDW + col];
        c = __builtin_amdgcn_wmma_f32_16x16x4_f32(false, a, false, b, (short)0, c, false, false);
    }
    #pragma unroll
    for (int i = 0; i < 8; i++) {
        int n = n0 + i + kh * 8;
        if (n < N) xl[(size_t)n * HIDW + col] = c[i];
    }
}

// per-node attention scalars: a_src[n,h], a_dst[n,h]
__global__ void k_attnode(const float* __restrict__ xl, const float* __restrict__ att_src,
                          const float* __restrict__ att_dst, int l,
                          float* __restrict__ asrc, float* __restrict__ adst, int N) {
    int i = blockIdx.x * blockDim.x + threadIdx.x;   // n*4+h
    if (i >= N * HEADS) return;
    int n = i >> 2, hh = i & 3;
    const float* row = xl + (size_t)n * HIDW + hh * CHV;
    const float* as = att_src + (size_t)(l * HEADS + hh) * CHV;
    const float* ad = att_dst + (size_t)(l * HEADS + hh) * CHV;
    float s = 0.f, d = 0.f;
    #pragma unroll
    for (int c = 0; c < CHV; c++) { float v = row[c]; s += v * as[c]; d += v * ad[c]; }
    asrc[i] = s; adst[i] = d;
}

// pass 1: alpha_raw = leaky_relu(a_src[src]+a_dst[dst]+a_edge); atomic segment-max
__global__ void k_alpha1(const int* __restrict__ src, const int* __restrict__ dst,
                         const float* __restrict__ asrc, const float* __restrict__ adst,
                         const float* __restrict__ aedge_all, int l,
                         float* __restrict__ alphabuf, unsigned* __restrict__ amax, int E) {
    int e = blockIdx.x * blockDim.x + threadIdx.x;
    if (e >= E) return;
    int s = src[e], d = dst[e];
    #pragma unroll
    for (int hh = 0; hh < HEADS; hh++) {
        float v = asrc[s * 4 + hh] + adst[d * 4 + hh] + aedge_all[(size_t)e * 12 + l * 4 + hh];
        v = v > 0.f ? v : 0.2f * v;
        alphabuf[(size_t)e * 4 + hh] = v;
        atomicMax(&amax[d * 4 + hh], fkey(v));
    }
}

// pass 2: exponentiate, accumulate denominator
__global__ void k_alpha2(const int* __restrict__ dst, const unsigned* __restrict__ amax,
                         float* __restrict__ alphabuf, float* __restrict__ denom, int E) {
    int e = blockIdx.x * blockDim.x + threadIdx.x;
    if (e >= E) return;
    int d = dst[e];
    #pragma unroll
    for (int hh = 0; hh < HEADS; hh++) {
        float m = funkey(amax[d * 4 + hh]);
        float p = __expf(alphabuf[(size_t)e * 4 + hh] - m);
        alphabuf[(size_t)e * 4 + hh] = p;
        atomicAdd(&denom[d * 4 + hh], p);
    }
}

// pass 3: out[dst] += (alpha/denom[dst]) * xl[src]   (128 threads = one edge row)
__global__ __launch_bounds__(128)
void k_scatter(const int* __restrict__ src, const int* __restrict__ dst,
               const float* __restrict__ alphabuf, const float* __restrict__ denom,
               const float* __restrict__ xl, float* __restrict__ out, int E) {
    int t = threadIdx.x;          // 0..127 -> (h = t>>5, c = t&31)
    int hh = t >> 5;
    for (int e = blockIdx.x; e < E; e += gridDim.x) {
        int s = src[e], d = dst[e];
        float w = alphabuf[(size_t)e * 4 + hh] / denom[d * 4 + hh];
        atomicAdd(&out[(size_t)d * HIDW + t], xl[(size_t)s * HIDW + t] * w);
    }
}

// bias + relu in place
__global__ void k_finalize(float* __restrict__ out, const float* __restrict__ gat_bias,
                           int l, int total) {
    int i = blockIdx.x * blockDim.x + threadIdx.x;
    if (i >= total) return;
    float v = out[i] + gat_bias[l * HIDW + (i & (HIDW - 1))];
    out[i] = v > 0.f ? v : 0.f;
}

// mean-pool (batch is sorted: run-length flush)
__global__ __launch_bounds__(128)
void k_pool(const float* __restrict__ h, const int* __restrict__ batch,
            float* __restrict__ gsum, float* __restrict__ cnts, int N) {
    const int CHUNK = 512;
    int k = threadIdx.x;
    int start = blockIdx.x * CHUNK;
    if (start >= N) return;
    int end = start + CHUNK; if (end > N) end = N;
    float acc = 0.f, cnt = 0.f;
    int curg = batch[start];
    for (int n = start; n < end; n++) {
        int g = batch[n];
        if (g != curg) {
            atomicAdd(&gsum[curg * HIDW + k], acc); acc = 0.f;
            if (k == 0) { atomicAdd(&cnts[curg], cnt); cnt = 0.f; }
            curg = g;
        }
        acc += h[(size_t)n * HIDW + k];
        cnt += 1.f;
    }
    atomicAdd(&gsum[curg * HIDW + k], acc);
    if (k == 0) atomicAdd(&cnts[curg], cnt);
}

// tiny regressor head: out[g] = relu(mean @ W1 + b1) @ W2 + b2  (single block)
__global__ __launch_bounds__(128)
void k_mlp(const float* __restrict__ gsum, const float* __restrict__ cnts,
           const float* __restrict__ W1, const float* __restrict__ b1,
           const float* __restrict__ W2, const float* __restrict__ b2,
           float* __restrict__ outp) {
    __shared__ float gv[HIDW];
    __shared__ float red[HIDW];
    int t = threadIdx.x;
    for (int gi = 0; gi < GRAPHS; gi++) {
        float cn = cnts[gi]; cn = cn > 1.f ? cn : 1.f;
        gv[t] = gsum[gi * HIDW + t] / cn;
        __syncthreads();
        float s = b1[t];
        for (int k = 0; k < HIDW; k++) s += gv[k] * W1[k * HIDW + t];
        s = s > 0.f ? s : 0.f;
        red[t] = s * W2[t];
        __syncthreads();
        for (int off = 64; off > 0; off >>= 1) {
            if (t < off) red[t] += red[t + off];
            __syncthreads();
        }
        if (t == 0) outp[gi] = red[0] + b2[0];
        __syncthreads();
    }
}

// =====================================================================
extern "C" void kernel_launch(void* const* d_in, const int* in_sizes, int n_in,
                              void* d_out, int out_size, void* d_ws, size_t ws_size,
                              hipStream_t stream) {
    (void)n_in; (void)out_size; (void)ws_size;
    const float* x          = (const float*)d_in[0];
    const int*   eidx       = (const int*)  d_in[1];
    const float* edge_attr  = (const float*)d_in[2];
    const int*   batch      = (const int*)  d_in[3];
    const float* recipe     = (const float*)d_in[4];
    const float* W_in       = (const float*)d_in[5];
    const float* b_in       = (const float*)d_in[6];
    const float* W_edge     = (const float*)d_in[7];
    const float* b_edge     = (const float*)d_in[8];
    const float* lin_w      = (const float*)d_in[9];
    const float* lin_edge_w = (const float*)d_in[10];
    const float* att_src    = (const float*)d_in[11];
    const float* att_dst    = (const float*)d_in[12];
    const float* att_edge   = (const float*)d_in[13];
    const float* gat_bias   = (const float*)d_in[14];
    const float* W1         = (const float*)d_in[15];
    const float* b1         = (const float*)d_in[16];
    const float* W2         = (const float*)d_in[17];
    const float* b2         = (const float*)d_in[18];

    const int N = in_sizes[3];
    const int E = in_sizes[2] / EDGE_DIM;
    const int* src = eidx;
    const int* dst = eidx + E;

    // workspace layout
    float* ws = (float*)d_ws;
    float* h0      = ws;                         // N*128
    float* h1      = h0 + (size_t)N * HIDW;      // N*128
    float* xl      = h1 + (size_t)N * HIDW;      // N*128
    float* asrc    = xl + (size_t)N * HIDW;      // N*4
    float* adst    = asrc + (size_t)N * 4;       // N*4
    unsigned* amax = (unsigned*)(adst + (size_t)N * 4);  // N*4
    float* denom   = (float*)amax + (size_t)N * 4;       // N*4
    float* aedge   = denom + (size_t)N * 4;      // E*12
    float* alpha   = aedge + (size_t)E * 12;     // E*4
    float* wattall = alpha + (size_t)E * 4;      // 128*12
    float* wc      = wattall + HIDW * 12;        // 16*12
    float* bcomb   = wc + EDGE_DIM * 12;         // 12
    float* gsum    = bcomb + 12;                 // 8*128
    float* cnts    = gsum + GRAPHS * HIDW;       // 8

    // precompute folded edge-attention path (no E x 128 buffers ever materialized)
    k_attw<<<(HIDW * 12 + 255) / 256, 256, 0, stream>>>(lin_edge_w, att_edge, wattall);
    k_wc<<<1, 256, 0, stream>>>(W_edge, b_edge, wattall, wc, bcomb);
    k_aedge<<<(E + 255) / 256, 256, 0, stream>>>(edge_attr, wc, bcomb, aedge, E);

    // input projection (WMMA)
    k_inproj<<<(N + 15) / 16, 256, 0, stream>>>(x, batch, recipe, W_in, b_in, h0, N);

    hipMemsetAsync(gsum, 0, (GRAPHS * HIDW + GRAPHS) * sizeof(float), stream);

    float* hcur = h0;
    float* hnext = h1;
    for (int l = 0; l < LAYERS; l++) {
        k_lin<<<(N + 15) / 16, 256, 0, stream>>>(hcur, lin_w + (size_t)l * HIDW * HIDW, xl, N);
        k_attnode<<<(N * HEADS + 255) / 256, 256, 0, stream>>>(xl, att_src, att_dst, l, asrc, adst, N);
        hipMemsetAsync(amax, 0, (size_t)N * 4 * sizeof(unsigned), stream);
        hipMemsetAsync(denom, 0, (size_t)N * 4 * sizeof(float), stream);
        hipMemsetAsync(hnext, 0, (size_t)N * HIDW * sizeof(float), stream);
        k_alpha1<<<(E + 255) / 256, 256, 0, stream>>>(src, dst, asrc, adst, aedge, l, alpha, amax, E);
        k_alpha2<<<(E + 255) / 256, 256, 0, stream>>>(dst, amax, alpha, denom, E);
        k_scatter<<<16384, 128, 0, stream>>>(src, dst, alpha, denom, xl, hnext, E);
        k_finalize<<<((size_t)N * HIDW + 255) / 256, 256, 0, stream>>>(hnext, gat_bias, l, N * HIDW);
        float* tmp = hcur; hcur = hnext; hnext = tmp;
    }

    k_pool<<<(N + 511) / 512, 128, 0, stream>>>(hcur, batch, gsum, cnts, N);
    k_mlp<<<1, 128, 0, stream>>>(gsum, cnts, W1, b1, W2, b2, (float*)d_out);
}